// prms_marrmot_changed_53781580480881
// MI455X (gfx1250) — compile-verified
//
#include <hip/hip_runtime.h>
#include <math.h>

#define NB    2000
#define NT    365
#define NPAR  12
#define NMUL  8
#define LENF  15
#define NTILE 23   // ceil(365/16)

typedef __attribute__((ext_vector_type(2))) float v2f;
typedef __attribute__((ext_vector_type(8))) float v8f;

// ---------------------------------------------------------------------------
// Kernel 1: sequential 365-step scan. One thread per (basin, mul).
// States live in registers; Q = mean over the 8 mul lanes via shfl butterfly.
// Memory-bound: streams ~290 MB (params dominate) -> ~12.5 us floor @ 23.3 TB/s.
// ---------------------------------------------------------------------------
__global__ __launch_bounds__(256) void hydro_scan_kernel(
    const float* __restrict__ P, const float* __restrict__ Ep,
    const float* __restrict__ Tair, const float* __restrict__ params,
    float* __restrict__ Qout)
{
    int tid = blockIdx.x * blockDim.x + threadIdx.x;
    if (tid >= NB * NMUL) return;          // 16000 % 8 == 0: groups stay intact
    int b = tid >> 3;
    int m = tid & 7;

    float S1 = 0.5f, S2 = 0.5f, S3 = 0.5f, S4 = 0.5f, S5 = 0.5f;

    const float* __restrict__ pb = params + (size_t)b * NT * NPAR * NMUL + m;
    const float* __restrict__ Pb = P    + (size_t)b * NT;
    const float* __restrict__ Eb = Ep   + (size_t)b * NT;
    const float* __restrict__ Tb = Tair + (size_t)b * NT;

    for (int t = 0; t < NT; ++t) {
        float Pt  = Pb[t];
        float Ept = Eb[t];
        float Tt  = Tb[t];
        const float* __restrict__ pr = pb + (size_t)t * (NPAR * NMUL);
        // prefetch next step's parameter block (global_prefetch_b8)
        __builtin_prefetch(pr + NPAR * NMUL, 0, 1);

        float p0  = pr[0 * NMUL];
        float p1  = pr[1 * NMUL];
        float p3  = pr[3 * NMUL];
        float p4  = pr[4 * NMUL];
        float p5  = pr[5 * NMUL];
        float p6  = pr[6 * NMUL];
        float p8  = pr[8 * NMUL];
        float p9  = pr[9 * NMUL];
        float p10 = pr[10 * NMUL];

        float tt    = fmaf(p0, 8.0f, -3.0f);
        float ddf   = p1 * 20.0f;
        float Smax3 = fmaf(p3, 680.0f, 20.0f);
        float p_lo  = fmaf(p4, 0.99f, 0.005f);
        float p_hi  = fmaf(p5, 0.99f, 0.005f);
        float p_exp = fmaf(p6, 4.0f, 1.0f);
        float i1 = p8;
        float i2 = p9 * 0.001f;
        float kb = p10;

        float snow = (Tt <= tt) ? Pt : 0.0f;     // DT == 1.0 -> S/DT == S
        float rain = Pt - snow;
        float melt = fmaxf(fminf(ddf * (Tt - tt), S1), 0.0f);
        float S1n  = S1 + snow - melt;

        float inter = rain * 0.95f;
        float evap2 = fminf(S2, Ept);
        float S2n   = fmaxf(S2 + (rain - inter) - evap2, 0.0f);

        float infil  = melt + inter;
        float S3c    = fmaxf(S3, 0.0f);
        float frac   = S3c / Smax3;
        float satexc = (p_lo + (p_hi - p_lo) * frac) * infil;
        float rem    = fmaxf(infil - satexc, 0.0f);
        float rech   = rem * __powf(frac, p_exp);
        float evap3  = fminf(frac * Ept, S3);
        float S3n    = S3 + rem - rech - evap3;

        float S4c   = fmaxf(S4, 0.0f);
        float iflow = fminf(S4c, fmaf(i2 * S4c, S4c, i1 * S4c));
        float S4n   = S4 + rech - iflow;

        float base = kb * S5;
        float S5n  = S5 + iflow - base;

        S1 = S1n; S2 = S2n; S3 = S3n; S4 = S4n; S5 = S5n;

        // ensemble mean over the 8 mul lanes (aligned sub-groups of 8)
        float q = satexc + base;
        q += __shfl_xor(q, 1, 8);
        q += __shfl_xor(q, 2, 8);
        q += __shfl_xor(q, 4, 8);
        if (m == 0) Qout[(size_t)b * NT + t] = q * 0.125f;
    }
}

// ---------------------------------------------------------------------------
// Kernel 2: per-basin gamma routing weights.
// Note: exp(gammaln(aa)) * th^aa is K-independent and cancels in the
// normalization w /= w.sum(), so no lgamma needed.
// Stores 16 taps (tap 15 = 0) so the conv kernel gets K=16 for WMMA.
// ---------------------------------------------------------------------------
__global__ __launch_bounds__(128) void routing_weights_kernel(
    const float* __restrict__ rout_a, const float* __restrict__ rout_b,
    float* __restrict__ W)
{
    int b = blockIdx.x * blockDim.x + threadIdx.x;
    if (b >= NB) return;
    float aa = fmaxf(rout_a[b] * 2.9f, 0.0f) + 0.1f;
    float th = fmaxf(rout_b[b] * 6.5f, 0.0f) + 0.5f;
    float invth = 1.0f / th;
    float u[LENF];
    float s = 0.0f;
    #pragma unroll
    for (int k = 0; k < LENF; ++k) {
        float tk = (float)k + 0.5f;
        float v  = __powf(tk, aa - 1.0f) * __expf(-tk * invth);
        u[k] = v;
        s   += v;
    }
    float inv = 1.0f / s;
    #pragma unroll
    for (int k = 0; k < LENF; ++k) W[(size_t)b * 16 + k] = u[k] * inv;
    W[(size_t)b * 16 + 15] = 0.0f;
}

// ---------------------------------------------------------------------------
// Kernel 3: causal conv as fp32 WMMA. One wave per (basin, 16-timestep tile).
// D(16x16) = Toeplitz(Q windows)(16x16, as 4 chained K=4 chunks) x w(16x16,
// replicated across N). Only column N=0 is consumed (lanes 0 and 16 hold it).
// A layout (16x4 f32): lanes 0-15 -> {K=0,K=1}, lanes 16-31 -> {K=2,K=3}.
// All index masking is branchless (v_cndmask) so EXEC stays all-ones at WMMA.
// ---------------------------------------------------------------------------
__global__ __launch_bounds__(128) void routing_conv_wmma_kernel(
    const float* __restrict__ Q, const float* __restrict__ W,
    float* __restrict__ out)
{
    int wave = blockIdx.x * (blockDim.x >> 5) + (threadIdx.x >> 5);
    int lane = threadIdx.x & 31;
    if (wave >= NB * NTILE) return;        // uniform per wave

    int b    = wave / NTILE;
    int tile = wave - b * NTILE;
    int t0   = tile * 16;

    const float* __restrict__ Qb = Q + (size_t)b * NT;
    const float* __restrict__ Wb = W + (size_t)b * 16;

    int M    = lane & 15;                  // output-row index this lane feeds
    int koff = (lane >> 4) << 1;           // 0 for lanes 0-15, 2 for 16-31

    v8f acc = {};
    #pragma unroll
    for (int c = 0; c < 4; ++c) {
        int k0 = 4 * c + koff;
        int i0 = t0 + M - k0;              // Q time index for K = k0
        int i1 = i0 - 1;                   //                 K = k0 + 1
        int c0 = min(max(i0, 0), NT - 1);
        int c1 = min(max(i1, 0), NT - 1);
        float a0 = Qb[c0];
        float a1 = Qb[c1];
        a0 = (i0 >= 0 && i0 < NT) ? a0 : 0.0f;
        a1 = (i1 >= 0 && i1 < NT) ? a1 : 0.0f;
        v2f a  = {a0, a1};
        v2f bw = {Wb[k0], Wb[k0 + 1]};     // replicated across all N columns
        acc = __builtin_amdgcn_wmma_f32_16x16x4_f32(
                  /*neg_a=*/false, a, /*neg_b=*/false, bw,
                  /*c_mod=*/(short)0, acc, /*reuse_a=*/false, /*reuse_b=*/false);
    }

    // D column N=0: lane 0 holds rows 0..7 in acc[0..7]; lane 16 rows 8..15.
    if ((lane & 15) == 0) {
        int mbase = (lane >> 4) * 8;
        #pragma unroll
        for (int j = 0; j < 8; ++j) {
            int t = t0 + mbase + j;
            if (t < NT) out[(size_t)b * NT + t] = acc[j];
        }
    }
}

extern "C" void kernel_launch(void* const* d_in, const int* in_sizes, int n_in,
                              void* d_out, int out_size, void* d_ws, size_t ws_size,
                              hipStream_t stream)
{
    const float* P      = (const float*)d_in[0];
    const float* Ep     = (const float*)d_in[1];
    const float* Tair   = (const float*)d_in[2];
    const float* params = (const float*)d_in[3];
    const float* rout_a = (const float*)d_in[4];
    const float* rout_b = (const float*)d_in[5];
    float* out = (float*)d_out;

    // workspace layout: Q[NB*NT] then W[NB*16]  (~3.05 MB total)
    float* Qws = (float*)d_ws;
    float* Wws = Qws + (size_t)NB * NT;

    hydro_scan_kernel<<<(NB * NMUL + 255) / 256, 256, 0, stream>>>(
        P, Ep, Tair, params, Qws);
    routing_weights_kernel<<<(NB + 127) / 128, 128, 0, stream>>>(
        rout_a, rout_b, Wws);
    routing_conv_wmma_kernel<<<(NB * NTILE + 3) / 4, 128, 0, stream>>>(
        Qws, Wws, out);
}